// SentRNNDecoder_26826365730890
// MI455X (gfx1250) — compile-verified
//
#include <hip/hip_runtime.h>
#include <hip/hip_bf16.h>
#include <math.h>

// ---------------------------------------------------------------------------
// Pointer-network LSTM decoder for MI455X (gfx1250, wave32).
//   B=256, S=512, E=512, H=512, T=8 (fixed, matching the reference).
// GEMMs: V_WMMA_F32_16X16X4_F32 (fp32 matrix pipe, keeps reference numerics
// for the argmax feedback), staged through LDS with double-buffered
// GLOBAL_LOAD_ASYNC_TO_LDS_B128 (ASYNCcnt) so global traffic overlaps WMMA.
// Score pass is bandwidth-bound (256 MB context re-read per step) -> wide
// float4 loads + wave32 shuffle reduction.
// ---------------------------------------------------------------------------

typedef float v2f __attribute__((ext_vector_type(2)));
typedef float v8f __attribute__((ext_vector_type(8)));

#define B_ 256
#define S_ 512
#define E_ 512
#define H_ 512
#define T_ 8

// GEMM tiling: 128 threads = 4 waves. Block tile 64(M) x 64(N), BK=16.
// Wave w owns rows [16w,16w+16), 4 accumulators cover the 64 N columns.
#define BM 64
#define BN 64
#define BK 16

__device__ __forceinline__ void async_wait0() {
  asm volatile("s_wait_asynccnt 0x0" ::: "memory");
}

// Issue async global->LDS copies for one 64x16 f32 tile (256 float4 slots,
// 128 threads -> 2 b128 copies each). LDS layout uses a 16B-granule XOR
// swizzle (slot ^= (row>>2)&3) so the 16 rows read by a half-wave later hit
// distinct banks, with every access staying 16B/8B aligned (no padding).
__device__ __forceinline__ void stage_tile_async(
    const float* __restrict__ G, int K, int g_row0, int k0,
    unsigned lds_base_bytes, int tid)
{
#pragma unroll
  for (int i = 0; i < 2; ++i) {
    const int linear = tid + i * 128;           // 0..255
    const int row = linear >> 2;                // 4 float4 slots per row
    const int c4  = linear & 3;
    const int sl  = c4 ^ ((row >> 2) & 3);      // bank swizzle
    const unsigned long long gaddr =
        (unsigned long long)(G + (size_t)(g_row0 + row) * K + k0 + c4 * 4);
    const unsigned laddr =
        lds_base_bytes + (unsigned)((row * BK + sl * 4) * sizeof(float));
    asm volatile("global_load_async_to_lds_b128 %0, %1, off"
                 :: "v"(laddr), "v"(gaddr) : "memory");
  }
}

// Read two consecutive floats [row][col..col+1] honoring the XOR swizzle.
__device__ __forceinline__ float2 lds_frag(const float* buf, int row, int col) {
  const int sl = (col >> 2) ^ ((row >> 2) & 3);
  return *(const float2*)(buf + row * BK + sl * 4 + (col & 3));
}

// C[m,n] = sum_k A0[m,k]*W0[n,k] (+ sum_k A1[m,k]*W1[n,k]) + bias0[n] (+ bias1[n])
// A row-major [M,K], W row-major [N,K]  (i.e. C = A @ W^T, "NT" gemm).
__global__ __launch_bounds__(128) void wmma_gemm_nt(
    const float* __restrict__ A0, const float* __restrict__ W0, int K0,
    const float* __restrict__ A1, const float* __restrict__ W1, int K1,
    const float* __restrict__ bias0, const float* __restrict__ bias1,
    float* __restrict__ C, int N)
{
  __shared__ __align__(16) float Xs[2][BM * BK];
  __shared__ __align__(16) float Ws[2][BN * BK];

  const int tid  = threadIdx.x;
  const int wave = tid >> 5;       // wave32
  const int lane = tid & 31;
  const int half = (lane >> 4) & 1;
  const int l16  = lane & 15;
  const int m0 = blockIdx.y * BM;
  const int n0 = blockIdx.x * BN;

  const unsigned xb0 = (unsigned)(size_t)(&Xs[0][0]);
  const unsigned xb1 = (unsigned)(size_t)(&Xs[1][0]);
  const unsigned wb0 = (unsigned)(size_t)(&Ws[0][0]);
  const unsigned wb1 = (unsigned)(size_t)(&Ws[1][0]);

  v8f acc[4] = {};

  for (int pass = 0; pass < 2; ++pass) {
    const float* __restrict__ A = pass ? A1 : A0;
    const float* __restrict__ W = pass ? W1 : W0;
    const int K = pass ? K1 : K0;
    if (A == nullptr) continue;                 // uniform across the block

    const int ntiles = K / BK;

    // Prologue: async-stage tile 0 into buffer 0.
    stage_tile_async(A, K, m0, 0, xb0, tid);
    stage_tile_async(W, K, n0, 0, wb0, tid);
    async_wait0();
    __syncthreads();

    for (int kt = 0; kt < ntiles; ++kt) {
      const int cur = kt & 1;

      // Overlap: kick off tile kt+1 into the other buffer before computing.
      // (Safe: barrier at end of the previous iteration guarantees every
      //  wave finished reading that buffer.)
      if (kt + 1 < ntiles) {
        const int k0n = (kt + 1) * BK;
        stage_tile_async(A, K, m0, k0n, cur ? xb0 : xb1, tid);
        stage_tile_async(W, K, n0, k0n, cur ? wb0 : wb1, tid);
      }

      const float* Xb = &Xs[cur][0];
      const float* Wb = &Ws[cur][0];

      // Front-load all fragments for this BK tile (one dscnt wait),
      // then issue the 16 WMMAs back-to-back on 4 independent acc chains.
      // Layouts per ISA 7.12.2: A 16x4 f32 -> lanes 0-15 K=kk,kk+1; lanes
      // 16-31 K=kk+2,kk+3. B 4x16 symmetric. C/D: VGPR r -> M=r / M=8+r.
      v2f a4[4];
      v2f b44[4][4];
#pragma unroll
      for (int kk4 = 0; kk4 < 4; ++kk4) {
        const int col = kk4 * 4 + 2 * half;
        const float2 av = lds_frag(Xb, wave * 16 + l16, col);
        a4[kk4].x = av.x; a4[kk4].y = av.y;
#pragma unroll
        for (int j = 0; j < 4; ++j) {
          const float2 bv = lds_frag(Wb, j * 16 + l16, col);
          b44[kk4][j].x = bv.x; b44[kk4][j].y = bv.y;
        }
      }
#pragma unroll
      for (int kk4 = 0; kk4 < 4; ++kk4) {
#pragma unroll
        for (int j = 0; j < 4; ++j) {
          acc[j] = __builtin_amdgcn_wmma_f32_16x16x4_f32(
              /*neg_a=*/false, a4[kk4], /*neg_b=*/false, b44[kk4][j],
              /*c_mod=*/(short)0, acc[j], /*reuse_a=*/false, /*reuse_b=*/false);
        }
      }

      // Next tile's async data must have landed, and all waves must be done
      // reading the current buffer before it is overwritten.
      async_wait0();
      __syncthreads();
    }
  }

  // Epilogue: C/D layout -> VGPR r: lane<16 M=r, lane>=16 M=8+r; N=l16.
#pragma unroll
  for (int j = 0; j < 4; ++j) {
    const int col = n0 + j * 16 + l16;
    float bv = 0.0f;
    if (bias0) bv += bias0[col];
    if (bias1) bv += bias1[col];
#pragma unroll
    for (int r = 0; r < 8; ++r) {
      const int row = m0 + wave * 16 + r + 8 * half;
      C[(size_t)row * N + col] = acc[j][r] + bv;
    }
  }
}

// h0/c0 vectors: ones @ W^T + b  ==  rowsum(W) + b  (identical for every b).
__global__ __launch_bounds__(256) void init_vec(
    const float* __restrict__ Wh, const float* __restrict__ bh,
    const float* __restrict__ Wc, const float* __restrict__ bc,
    float* __restrict__ h0v, float* __restrict__ c0v)
{
  const int j = blockIdx.x * 256 + threadIdx.x;   // 0..H-1
  float sh = 0.0f, sc = 0.0f;
  for (int k = 0; k < H_; k += 4) {
    const float4 a = *(const float4*)(Wh + (size_t)j * H_ + k);
    const float4 b = *(const float4*)(Wc + (size_t)j * H_ + k);
    sh += a.x + a.y + a.z + a.w;
    sc += b.x + b.y + b.z + b.w;
  }
  h0v[j] = sh + bh[j];
  c0v[j] = sc + bc[j];
}

__global__ __launch_bounds__(256) void bcast_state(
    const float* __restrict__ h0v, const float* __restrict__ c0v,
    float* __restrict__ h, float* __restrict__ c)
{
  const int idx = blockIdx.x * 256 + threadIdx.x;  // 0..B*H-1
  const int j = idx & (H_ - 1);
  h[idx] = h0v[j];
  c[idx] = c0v[j];
}

// gates [B,4H] (order i,f,g,o) -> updates h,c in place, mirrors to hs/cs outputs.
__global__ __launch_bounds__(256) void lstm_pointwise(
    const float* __restrict__ gates, float* __restrict__ h, float* __restrict__ c,
    float* __restrict__ hs_out, float* __restrict__ cs_out)
{
  const int idx = blockIdx.x * 256 + threadIdx.x;  // 0..B*H-1
  const int b = idx >> 9;                          // /H_
  const int j = idx & (H_ - 1);
  const float* g = gates + (size_t)b * 4 * H_;
  const float gi = g[j];
  const float gf = g[H_ + j];
  const float gg = g[2 * H_ + j];
  const float go = g[3 * H_ + j];
  const float si = 1.0f / (1.0f + __expf(-gi));
  const float sf = 1.0f / (1.0f + __expf(-gf));
  const float so = 1.0f / (1.0f + __expf(-go));
  const float c2 = sf * c[idx] + si * tanhf(gg);
  const float h2 = so * tanhf(c2);
  c[idx] = c2;
  h[idx] = h2;
  hs_out[idx] = h2;
  cs_out[idx] = c2;
}

// pen[b,s] = mask>0 ? context[b,s,:].q[b,:] : -1e9.  One wave per (b,s) row:
// lanes stream E=512 floats as float4 (coalesced b128 loads), wave reduce.
__global__ __launch_bounds__(256) void score_kernel(
    const float* __restrict__ ctx, const float* __restrict__ mask,
    const float* __restrict__ q, float* __restrict__ pen_out)
{
  const int wave = threadIdx.x >> 5;
  const int lane = threadIdx.x & 31;
  const int row = blockIdx.x * 8 + wave;           // 0..B*S-1
  const int b = row >> 9;                          // /S_
  const int s = row & (S_ - 1);
  const float* cr = ctx + ((size_t)b * S_ + s) * E_;
  const float* qr = q + (size_t)b * E_;
  float acc = 0.0f;
#pragma unroll
  for (int it = 0; it < E_ / 128; ++it) {
    const int k = it * 128 + lane * 4;
    const float4 cv = *(const float4*)(cr + k);
    const float4 qv = *(const float4*)(qr + k);
    acc += cv.x * qv.x + cv.y * qv.y + cv.z * qv.z + cv.w * qv.w;
  }
#pragma unroll
  for (int off = 16; off > 0; off >>= 1)
    acc += __shfl_down(acc, off, 32);
  if (lane == 0) {
    const float pen = (mask[(size_t)b * S_ + s] > 0.0f) ? acc : -1e9f;
    pen_out[(size_t)b * S_ + s] = pen;
  }
}

// Per-batch argmax over S (first-max tie-break like jnp.argmax), writes the
// pointer logit and gathers the chosen context row as the next decoder input.
__global__ __launch_bounds__(256) void argmax_gather(
    const float* __restrict__ pen, const float* __restrict__ ctx,
    float* __restrict__ logits_t, float* __restrict__ x)
{
  __shared__ float sv[256];
  __shared__ int   si[256];
  __shared__ int   sidx;
  const int b = blockIdx.x;
  const int tid = threadIdx.x;
  const float* p = pen + (size_t)b * S_;

  float bv = p[tid];
  int   bi = tid;
  const float v2 = p[tid + 256];
  if (v2 > bv) { bv = v2; bi = tid + 256; }        // keep lower index on tie
  sv[tid] = bv; si[tid] = bi;
  __syncthreads();
  for (int off = 128; off > 0; off >>= 1) {
    if (tid < off) {
      const float ov = sv[tid + off];
      const int   oi = si[tid + off];
      if (ov > sv[tid] || (ov == sv[tid] && oi < si[tid])) {
        sv[tid] = ov; si[tid] = oi;
      }
    }
    __syncthreads();
  }
  if (tid == 0) { sidx = si[0]; logits_t[b] = (float)si[0]; }
  __syncthreads();
  const int top = sidx;
  const float* crow = ctx + ((size_t)b * S_ + top) * E_;
  x[(size_t)b * E_ + tid]       = crow[tid];
  x[(size_t)b * E_ + tid + 256] = crow[tid + 256];
}

extern "C" void kernel_launch(void* const* d_in, const int* in_sizes, int n_in,
                              void* d_out, int out_size, void* d_ws, size_t ws_size,
                              hipStream_t stream)
{
  const float* context    = (const float*)d_in[0];
  const float* mask       = (const float*)d_in[1];
  const float* last_state = (const float*)d_in[2];
  const float* W_ih       = (const float*)d_in[3];
  const float* W_hh       = (const float*)d_in[4];
  const float* b_ih       = (const float*)d_in[5];
  const float* b_hh       = (const float*)d_in[6];
  const float* W_init_h   = (const float*)d_in[7];
  const float* b_init_h   = (const float*)d_in[8];
  const float* W_init_c   = (const float*)d_in[9];
  const float* b_init_c   = (const float*)d_in[10];
  const float* W_attn     = (const float*)d_in[11];
  const float* b_attn     = (const float*)d_in[12];

  // Outputs, concatenated in reference return order.
  float* out    = (float*)d_out;
  float* logits = out;                                  // [T,B]
  float* scores = out + (size_t)T_ * B_;                // [T,B,S]
  float* hs     = scores + (size_t)T_ * B_ * S_;        // [T,B,H]
  float* cs     = hs + (size_t)T_ * B_ * H_;            // [T,B,H]

  // Workspace carve-out (~4.2 MB of floats).
  float* ws    = (float*)d_ws;
  float* x     = ws;  ws += (size_t)B_ * E_;
  float* h     = ws;  ws += (size_t)B_ * H_;
  float* c     = ws;  ws += (size_t)B_ * H_;
  float* gates = ws;  ws += (size_t)B_ * 4 * H_;
  float* q     = ws;  ws += (size_t)B_ * E_;
  float* h0v   = ws;  ws += H_;
  float* c0v   = ws;  ws += H_;

  hipMemcpyAsync(x, last_state, sizeof(float) * (size_t)B_ * E_,
                 hipMemcpyDeviceToDevice, stream);
  init_vec<<<H_ / 256, 256, 0, stream>>>(W_init_h, b_init_h, W_init_c, b_init_c,
                                         h0v, c0v);
  bcast_state<<<(B_ * H_) / 256, 256, 0, stream>>>(h0v, c0v, h, c);

  for (int t = 0; t < T_; ++t) {
    // gates = x@W_ih^T + h@W_hh^T + b_ih + b_hh   (M=256, N=2048, K=512+512)
    dim3 ggrid(4 * H_ / BN, B_ / BM);
    wmma_gemm_nt<<<ggrid, 128, 0, stream>>>(x, W_ih, E_, h, W_hh, H_,
                                            b_ih, b_hh, gates, 4 * H_);
    lstm_pointwise<<<(B_ * H_) / 256, 256, 0, stream>>>(
        gates, h, c, hs + (size_t)t * B_ * H_, cs + (size_t)t * B_ * H_);

    // q = h@W_attn^T + b_attn   (M=256, N=512, K=512)
    dim3 agrid(E_ / BN, B_ / BM);
    wmma_gemm_nt<<<agrid, 128, 0, stream>>>(h, W_attn, H_, nullptr, nullptr, 0,
                                            b_attn, nullptr, q, E_);

    score_kernel<<<(B_ * S_) / 8, 256, 0, stream>>>(
        context, mask, q, scores + (size_t)t * B_ * S_);
    argmax_gather<<<B_, 256, 0, stream>>>(
        scores + (size_t)t * B_ * S_, context, logits + (size_t)t * B_, x);
  }
}